// InterpLnr_44976897524415
// MI455X (gfx1250) — compile-verified
//
#include <hip/hip_runtime.h>
#include <cstdint>
#include <cstddef>

// ---- problem constants (from reference) ----
#define NSEG   7            // MAX_NUM_SEG = 128//19 + 1
#define SGRID  64           // SEG_GRID = 2*MAX_LEN_SEG
#define GTOT   (NSEG*SGRID) // 448 grid points per batch row
#define TPAD   192          // MAX_LEN_PAD
#define CDIM   80
#define C4     (CDIM/4)     // 20 float4 per row
#define ROWS   128          // rows of x actually reachable (idx_org <= 126, +1)
#define STAGE  (ROWS*C4)    // 2560 float4 = 40 KB staged in LDS

typedef uint32_t u32;
typedef uint64_t u64;
typedef __attribute__((ext_vector_type(4))) float f32x4;
typedef __attribute__((ext_vector_type(4))) u32 u32x4_t;
typedef __attribute__((ext_vector_type(8))) int i32x8_t;
typedef __attribute__((ext_vector_type(4))) int i32x4_t;

#ifdef __HIP_DEVICE_COMPILE__
#  if __has_builtin(__builtin_amdgcn_tensor_load_to_lds)
#    define HAVE_TDM 1
#  endif
#  if __has_builtin(__builtin_amdgcn_s_wait_tensorcnt)
#    define HAVE_WAIT_TENSOR 1
#  endif
#endif
#ifndef HAVE_TDM
#  define HAVE_TDM 0
#endif
#ifndef HAVE_WAIT_TENSOR
#  define HAVE_WAIT_TENSOR 0
#endif

__global__ __launch_bounds__(256)
void InterpLnr_44976897524415_kernel(const float* __restrict__ x,
                                     const int*   __restrict__ len_seq,
                                     const float* __restrict__ scales,
                                     const int*   __restrict__ len_seg_raw,
                                     float*       __restrict__ out)
{
    __shared__ f32x4  s_x[STAGE];      // staged x[b, 0:128, :]   (40 KB)
    __shared__ int    s_len[NSEG];
    __shared__ int    s_off[NSEG];     // exclusive cumsum of len_seg
    __shared__ int    s_cnt[NSEG];     // valid grid points per segment
    __shared__ int    s_pre[NSEG];     // exclusive cumsum of s_cnt
    __shared__ int    s_src[TPAD];     // inverse map: out row -> src row (-1 = zero)
    __shared__ float  s_lam[TPAD];     // interpolation weight per out row
    __shared__ int    s_i0[GTOT];      // per-grid-point source row
    __shared__ float  s_gl[GTOT];      // per-grid-point lambda

    const int b   = blockIdx.x;
    const int tid = threadIdx.x;
    const float* __restrict__ xb = x + (size_t)b * (TPAD * CDIM);

    if (tid < NSEG) {
        s_len[tid] = len_seg_raw[b * NSEG + tid] + 19;  // + MIN_LEN_SEG
        s_cnt[tid] = 0;
    }
    if (tid < TPAD) s_src[tid] = -1;

    // ---- Stage x[b, 0:128, :] (40 KB, contiguous) into LDS via the TDM ----
    // One D#: 8-byte elements, tile 5120x1, stride 5120. Issued by wave 0 only
    // (TDM ignores EXEC; the branch keeps other waves from re-issuing it).
#if HAVE_TDM
    if (tid < 32) {
        const u32 ldsa = (u32)(uintptr_t)(void*)&s_x[0];      // low 32 bits = LDS offset
        const u64 ga   = (u64)(uintptr_t)xb;
        u32x4_t g0 = {0u, 0u, 0u, 0u};
        g0[0] = 1u;                                           // count=1 valid descriptor
        g0[1] = ldsa;                                         // lds_addr
        g0[2] = (u32)ga;                                      // global_addr[31:0]
        g0[3] = ((u32)(ga >> 32) & 0x01FFFFFFu) | 0x80000000u;// global_addr[56:32] | type=2
        i32x8_t g1 = {
            0x00030000,   // workgroup_mask=0, data_size=3 (8B elements)
            0x14000000,   // tensor_dim0 = 5120  (low 16 in bits 63:48)
            0x00010000,   // tensor_dim0 hi = 0, tensor_dim1 = 1
            0x14000000,   // tile_dim0 = 5120
            0x00000001,   // tile_dim1 = 1, tile_dim2 = 0
            5120,         // tensor_dim0_stride[31:0]
            0x14000000,   // stride0 hi = 0, tensor_dim1_stride lo16 = 5120
            0             // tensor_dim1_stride hi
        };
        i32x4_t g2 = {0, 0, 0, 0};
        i32x4_t g3 = {0, 0, 0, 0};
#if defined(__clang_major__) && __clang_major__ >= 23
        i32x8_t g4 = {0, 0, 0, 0, 0, 0, 0, 0};
        __builtin_amdgcn_tensor_load_to_lds(g0, g1, g2, g3, g4, 0);
#else
        __builtin_amdgcn_tensor_load_to_lds(g0, g1, g2, g3, 0);
#endif
    }
#else
    for (int e = tid; e < STAGE; e += 256)
        s_x[e] = reinterpret_cast<const f32x4*>(xb)[e];
#endif

    __syncthreads();

    // exclusive cumsum of segment lengths (7 elements, serial is fine)
    if (tid == 0) {
        int o = 0;
        for (int s = 0; s < NSEG; ++s) { s_off[s] = o; o += s_len[s]; }
    }
    __syncthreads();

    const int lq = len_seq[b];

    // ---- Per-grid-point resample math. floor(i/scale) is monotone in i, so
    // the combined mask is a prefix per segment: count = max valid i + 1. ----
    for (int g = tid; g < GTOT; g += 256) {
        const int   s   = g >> 6;
        const int   i   = g & 63;
        const float sc  = scales[b * NSEG + s] + 0.5f;
        const float isc = (float)i / sc;
        const float ifl = floorf(isc);
        const float lam = isc - ifl;
        const float iorg = ifl + (float)s_off[s];
        const bool  m = (ifl < (float)(s_len[s] - 1)) && (iorg < (float)(lq - 1));
        int i0 = (int)iorg;
        i0 = min(max(i0, 0), ROWS - 2);       // valid entries are always <= 126
        s_i0[g] = i0;
        s_gl[g] = lam;
        if (m) atomicMax(&s_cnt[s], i + 1);
    }
    __syncthreads();

    if (tid == 0) {
        int p = 0;
        for (int s = 0; s < NSEG; ++s) { s_pre[s] = p; p += s_cnt[s]; }
    }
    __syncthreads();

    // ---- Compaction: pos = pre[s] + i for the prefix-valid points ----
    for (int g = tid; g < GTOT; g += 256) {
        const int s = g >> 6;
        const int i = g & 63;
        if (i < s_cnt[s]) {
            const int pos = s_pre[s] + i;
            if (pos < TPAD) { s_src[pos] = s_i0[g]; s_lam[pos] = s_gl[g]; }
        }
    }

    // wave 0 makes sure the TDM transfer has landed before anyone reads s_x
#if HAVE_TDM
    if (tid < 32) {
#if HAVE_WAIT_TENSOR
        __builtin_amdgcn_s_wait_tensorcnt(0);
#else
        asm volatile("s_wait_tensorcnt 0x0" ::: "memory");
#endif
    }
#endif
    __syncthreads();

    // ---- Emit output: 192 rows x 20 float4, zeros where unmapped.
    // Streaming, never re-read -> non-temporal 128-bit stores. ----
    f32x4* __restrict__ ob = reinterpret_cast<f32x4*>(out) + (size_t)b * (TPAD * C4);
    for (int e = tid; e < TPAD * C4; e += 256) {
        const int p = e / C4;
        const int c = e - p * C4;
        const int src = s_src[p];
        f32x4 r;
        if (src >= 0) {
            const float lam = s_lam[p];
            const float w0  = 1.0f - lam;
            const f32x4 a   = s_x[src * C4 + c];
            const f32x4 b2  = s_x[src * C4 + C4 + c];
            r = w0 * a + lam * b2;
        } else {
            r = (f32x4){0.0f, 0.0f, 0.0f, 0.0f};
        }
        __builtin_nontemporal_store(r, ob + e);
    }
}

extern "C" void kernel_launch(void* const* d_in, const int* in_sizes, int n_in,
                              void* d_out, int out_size, void* d_ws, size_t ws_size,
                              hipStream_t stream) {
    (void)n_in; (void)out_size; (void)d_ws; (void)ws_size;
    const float* x           = (const float*)d_in[0];
    const int*   len_seq     = (const int*)d_in[1];
    const float* scales      = (const float*)d_in[2];
    const int*   len_seg_raw = (const int*)d_in[3];
    float*       out         = (float*)d_out;
    const int Bsz = in_sizes[1];   // len_seq has B elements
    InterpLnr_44976897524415_kernel<<<dim3(Bsz), dim3(256), 0, stream>>>(
        x, len_seq, scales, len_seg_raw, out);
}